// _ProposalTargetLayer_MSDN_23742579212566
// MI455X (gfx1250) — compile-verified
//
#include <hip/hip_runtime.h>
#include <cstdint>

// ---------------- problem constants (from reference) ----------------
#define B_    16
#define N_    12000
#define G_    128
#define M_    (N_ + G_)     // 12128 rois (proposals + appended gt)
#define R_    256
#define FGQ   64
#define NPAD  16384         // next pow2 >= M_, for bitonic sort

typedef float v2f __attribute__((ext_vector_type(2)));
typedef float v8f __attribute__((ext_vector_type(8)));
typedef unsigned int v4u __attribute__((ext_vector_type(4)));
typedef int v4i __attribute__((ext_vector_type(4)));
typedef int v8i __attribute__((ext_vector_type(8)));
typedef unsigned long long u64;

// ---------------- threefry2x32-20 (mirrors JAX PRNG structure) -------
__device__ __forceinline__ void tf2x32(unsigned k0, unsigned k1,
                                       unsigned c0, unsigned c1,
                                       unsigned& o0, unsigned& o1) {
  unsigned ks2 = 0x1BD11BDAu ^ k0 ^ k1;
  unsigned x0 = c0 + k0, x1 = c1 + k1;
  auto RND = [&](unsigned r) {
    x0 += x1;
    x1 = (x1 << r) | (x1 >> (32u - r));
    x1 ^= x0;
  };
  RND(13); RND(15); RND(26); RND(6);  x0 += k1;  x1 += ks2 + 1u;
  RND(17); RND(29); RND(16); RND(24); x0 += ks2; x1 += k0  + 2u;
  RND(13); RND(15); RND(26); RND(6);  x0 += k0;  x1 += k1  + 3u;
  RND(17); RND(29); RND(16); RND(24); x0 += k1;  x1 += ks2 + 4u;
  RND(13); RND(15); RND(26); RND(6);  x0 += ks2; x1 += k0  + 5u;
  o0 = x0; o1 = x1;
}

__device__ __forceinline__ float tf_uniform(unsigned k0, unsigned k1, unsigned ctr) {
  unsigned o0, o1;
  tf2x32(k0, k1, 0u, ctr, o0, o1);
  return (float)(o0 >> 8) * (1.0f / 16777216.0f);   // [0,1)
}

// ======================================================================
// Kernel 0: zero the per-batch fg/bg counters
// ======================================================================
__global__ void k_zero(int* __restrict__ p, int n) {
  int i = blockIdx.x * blockDim.x + threadIdx.x;
  if (i < n) p[i] = 0;
}

// ======================================================================
// Kernel 1: fused IoU max/argmax.
//   grid = (ceil(M/128), B), block = 256 (8 waves of 32).
//   GT tile (128x5 f32 = 2560B) staged in LDS via TDM tensor_load_to_lds.
//   Each wave owns a 16(roi) x 16(gt) tile; the union denominator
//   area_m + area_g is an outer sum computed by V_WMMA_F32_16X16X4_F32
//   whose C/D layout (lane<16: M=r,N=lane; lane>=16: M=r+8,N=lane-16)
//   directly feeds the per-lane VALU IoU. The argmax over g is tracked as
//   a fraction (bestn/bestd) via cross-multiplication (positive denoms),
//   so the expensive exact f32 divide happens once per roi, not per (m,g).
//   The 8-chunk loop is fully unrolled: gpre[] becomes direct registers
//   (no M0/MOVREL) and 8 static WMMA sites schedule against the ds loads.
// ======================================================================
__global__ __launch_bounds__(256) void k_iou(const float* __restrict__ rois,
                                             const float* __restrict__ gt,
                                             float* __restrict__ maxov,
                                             int* __restrict__ asgn) {
  __shared__ float gtraw[G_ * 5];      // raw gt rows [x1,y1,x2,y2,label]
  __shared__ float garea[G_];
  __shared__ float rc[8][16][4];       // per-wave roi coords
  __shared__ float rar[8][16];         // per-wave roi areas

  const int b    = blockIdx.y;
  const int tid  = threadIdx.x;
  const int wave = tid >> 5;
  const int lane = tid & 31;           // wave32
  const int half = lane >> 4;
  const int nin  = lane & 15;

  // ---- stage GT tile into LDS via Tensor Data Mover -------------------
#if __has_builtin(__builtin_amdgcn_tensor_load_to_lds)
  if (tid == 0) {
    uint64_t ga  = (uint64_t)(uintptr_t)(gt + (size_t)b * G_ * 5);
    unsigned lds = (unsigned)(uintptr_t)(&gtraw[0]);   // low 32b of flat = LDS offset
    v4u g0;
    g0.x = 1u;                                        // count=1, is_restore=0, no gather
    g0.y = lds;                                       // lds_addr [63:32]
    g0.z = (unsigned)ga;                              // global_addr [95:64]
    g0.w = ((unsigned)(ga >> 32) & 0x01FFFFFFu) | (2u << 30);  // addr[56:32] | type=2
    v8i g1;
    g1[0] = 0x00020000;                    // workgroup_mask=0, data_size=2 (4B)
    g1[1] = (int)(640u << 16);             // tensor_dim0 = 640 (low 16 @ [31:16])
    g1[2] = (int)(1u   << 16);             // tensor_dim0 hi=0, tensor_dim1 = 1
    g1[3] = (int)(640u << 16);             // tensor_dim1 hi=0, tile_dim0 = 640
    g1[4] = 1;                             // tile_dim1 = 1, tile_dim2 = 0
    g1[5] = 640;                           // tensor_dim0_stride (low 32)
    g1[6] = (int)(640u << 16);             // stride0 hi=0, tensor_dim1_stride low16
    g1[7] = 0;
    v4i z4 = {0, 0, 0, 0};
#if defined(__clang_major__) && __clang_major__ >= 23
    v8i z8 = {0, 0, 0, 0, 0, 0, 0, 0};
    __builtin_amdgcn_tensor_load_to_lds(g0, g1, z4, z4, z8, 0);
#else
    __builtin_amdgcn_tensor_load_to_lds(g0, g1, z4, z4, 0);
#endif
  }
  __builtin_amdgcn_s_wait_tensorcnt((short)0);
#else
  for (int i = tid; i < G_ * 5; i += 256) gtraw[i] = gt[(size_t)b * G_ * 5 + i];
#endif
  __syncthreads();

  // ---- derived LDS state: gt areas + per-wave roi staging -------------
  if (tid < G_) {
    float x1 = gtraw[tid * 5 + 0], y1 = gtraw[tid * 5 + 1];
    float x2 = gtraw[tid * 5 + 2], y2 = gtraw[tid * 5 + 3];
    garea[tid] = (x2 - x1 + 1.f) * (y2 - y1 + 1.f);
  }
  {
    int m = blockIdx.x * 128 + wave * 16 + nin;
    float x1 = 0.f, y1 = 0.f, x2 = -1.f, y2 = -1.f;   // dummy: zero area, zero IoU
    if (lane < 16 && m < M_) {
      if (m < N_) {
        const float* p = rois + ((size_t)b * N_ + m) * 5;
        x1 = p[1]; y1 = p[2]; x2 = p[3]; y2 = p[4];
      } else {
        const float* p = gt + ((size_t)b * G_ + (m - N_)) * 5;
        x1 = p[0]; y1 = p[1]; x2 = p[2]; y2 = p[3];
      }
    }
    if (lane < 16) {
      rc[wave][nin][0] = x1; rc[wave][nin][1] = y1;
      rc[wave][nin][2] = x2; rc[wave][nin][3] = y2;
      rar[wave][nin]   = (x2 - x1 + 1.f) * (y2 - y1 + 1.f);
    }
  }
  __syncthreads();

  // best IoU tracked as fraction bestn/bestd (bestd > 0 after first chunk)
  float bestn[8], bestd[8];
  int   bestg[8];
#pragma unroll
  for (int r = 0; r < 8; ++r) { bestn[r] = -1.f; bestd[r] = 1.f; bestg[r] = 0; }

  // A matrix 16x4 f32: K0 = roi area, K1 = 1, K2=K3 = 0
  v2f A;
  A.x = (lane < 16) ? rar[wave][nin] : 0.f;
  A.y = (lane < 16) ? 1.f            : 0.f;

  // preload per-chunk B-operand areas so the WMMA only waits on coord loads
  float gpre[8];
#pragma unroll
  for (int c = 0; c < 8; ++c)
    gpre[c] = (lane < 16) ? garea[c * 16 + nin] : 0.f;

#pragma unroll
  for (int c = 0; c < 8; ++c) {           // 8 chunks of 16 gts (fully unrolled)
    int g = c * 16 + nin;
    float gx1 = gtraw[g * 5 + 0], gy1 = gtraw[g * 5 + 1];
    float gx2 = gtraw[g * 5 + 2], gy2 = gtraw[g * 5 + 3];

    v8f D;
#if __has_builtin(__builtin_amdgcn_wmma_f32_16x16x4_f32)
    // B matrix 4x16 f32: row K0 = ones, row K1 = gt area, rows K2,K3 = 0
    v2f Bv;
    Bv.x = (lane < 16) ? 1.f : 0.f;
    Bv.y = gpre[c];
    v8f C = {0.f, 0.f, 0.f, 0.f, 0.f, 0.f, 0.f, 0.f};
    D = __builtin_amdgcn_wmma_f32_16x16x4_f32(false, A, false, Bv,
                                              (short)0, C, false, false);
#else
#pragma unroll
    for (int r = 0; r < 8; ++r) D[r] = rar[wave][r + half * 8] + gpre[c];
#endif

#pragma unroll
    for (int r = 0; r < 8; ++r) {
      int mloc = r + half * 8;
      float x1 = fmaxf(rc[wave][mloc][0], gx1);
      float y1 = fmaxf(rc[wave][mloc][1], gy1);
      float x2 = fminf(rc[wave][mloc][2], gx2);
      float y2 = fminf(rc[wave][mloc][3], gy2);
      float iw = fmaxf(x2 - x1 + 1.f, 0.f);
      float ih = fmaxf(y2 - y1 + 1.f, 0.f);
      float inter = iw * ih;
      float den   = D[r] - inter;          // union = area_m + area_g - inter
      // inter/den > bestn/bestd  <=>  inter*bestd > bestn*den  (den,bestd > 0)
      if (inter * bestd[r] > bestn[r] * den) {
        bestn[r] = inter; bestd[r] = den; bestg[r] = g;
      }
    }
  }

  // reduce across the 16 lanes of each half (ties -> lowest gt index)
#pragma unroll
  for (int r = 0; r < 8; ++r) {
    float n = bestn[r], d = bestd[r];
    int   g = bestg[r];
    for (int off = 1; off < 16; off <<= 1) {
      float n2 = __shfl_xor(n, off, 32);
      float d2 = __shfl_xor(d, off, 32);
      int   g2 = __shfl_xor(g, off, 32);
      float ca = n2 * d;      // candidate
      float cb = n * d2;      // incumbent
      if (ca > cb || (ca == cb && g2 < g)) { n = n2; d = d2; g = g2; }
    }
    if (nin == 0) {
      int m = blockIdx.x * 128 + wave * 16 + r + half * 8;
      if (m < M_) {
        maxov[(size_t)b * M_ + m] = n / d;   // single exact divide per roi
        asgn [(size_t)b * M_ + m] = g;
      }
    }
  }
}

// ======================================================================
// Kernel 2: fg/bg masks, threefry keys, packed sortable 64-bit keys,
//           per-batch fg/bg counts.  grid = (NPAD/256, B), block 256.
// ======================================================================
__global__ __launch_bounds__(256) void k_keys(const float* __restrict__ maxov,
                                              u64* __restrict__ keyfg,
                                              u64* __restrict__ keybg,
                                              int* __restrict__ fgnum,
                                              int* __restrict__ bgnum) {
  int b = blockIdx.y;
  int m = blockIdx.x * blockDim.x + threadIdx.x;   // 0..NPAD-1
  size_t o = (size_t)b * NPAD + m;
  if (m >= M_) { keyfg[o] = 0ull; keybg[o] = 0ull; return; }

  float ov = maxov[(size_t)b * M_ + m];
  bool fg = ov >= 0.5f;
  bool bg = (ov < 0.5f) && (ov >= 0.0f);

  // subkeys of jax.random.split(jax.random.key(42)) -- constant-folded
  unsigned a0, a1, b0, b1;
  tf2x32(0u, 42u, 0u, 1u, a0, a1);
  tf2x32(0u, 42u, 2u, 3u, b0, b1);
  unsigned ctr = (unsigned)(b * M_ + m);
  float r1 = tf_uniform(a0, a1, ctr);
  float r2 = tf_uniform(b0, b1, ctr);
  float kf = fg ? r1 + 2.f : r1;
  float kb = bg ? r2 + 2.f : r2;

  // positive floats: bit pattern is order-preserving. Low word ~m so that
  // ascending sort + read-from-end yields lowest index first on ties.
  keyfg[o] = ((u64)__float_as_uint(kf) << 32) | (u64)(0xFFFFFFFFu - (unsigned)m);
  keybg[o] = ((u64)__float_as_uint(kb) << 32) | (u64)(0xFFFFFFFFu - (unsigned)m);

  if (fg) atomicAdd(&fgnum[b], 1);
  if (bg) atomicAdd(&bgnum[b], 1);
}

// ======================================================================
// Kernel 3: per-batch bitonic sort of 16384 u64 keys in 128KB LDS
//           (possible thanks to CDNA5's 320KB WGP LDS). Writes top-R
//           indices in descending key order.  grid=(1,B), block=1024.
// ======================================================================
__global__ __launch_bounds__(1024) void k_sort(const u64* __restrict__ keys,
                                               int* __restrict__ top) {
  extern __shared__ u64 s[];
  int b = blockIdx.y;
  const u64* kin = keys + (size_t)b * NPAD;
  for (int i = threadIdx.x; i < NPAD; i += blockDim.x) s[i] = kin[i];

  for (int k = 2; k <= NPAD; k <<= 1) {
    for (int j = k >> 1; j > 0; j >>= 1) {
      __syncthreads();
      for (int i = threadIdx.x; i < NPAD; i += blockDim.x) {
        int ixj = i ^ j;
        if (ixj > i) {
          u64 a = s[i], c = s[ixj];
          bool up = ((i & k) == 0);           // ascending overall
          if (up ? (a > c) : (a < c)) { s[i] = c; s[ixj] = a; }
        }
      }
    }
  }
  __syncthreads();
  if (threadIdx.x < R_) {
    u64 key = s[NPAD - 1 - threadIdx.x];      // descending from the end
    top[b * R_ + threadIdx.x] = (int)(0xFFFFFFFFu - (unsigned)(key & 0xFFFFFFFFu));
  }
}

// ======================================================================
// Kernel 4: sample selection + bbox_transform + output writes.
//           grid = B, block = R.
// ======================================================================
__global__ __launch_bounds__(256) void k_gather(const float* __restrict__ rois,
                                                const float* __restrict__ gt,
                                                const int* __restrict__ asgn,
                                                const int* __restrict__ fgnum,
                                                const int* __restrict__ bgnum,
                                                const int* __restrict__ fgtop,
                                                const int* __restrict__ bgtop,
                                                float* __restrict__ out) {
  int b = blockIdx.x;
  int t = threadIdx.x;                       // 0..R-1
  int fgc = min(fgnum[b], FGQ);
  int bga = min(max(bgnum[b], 1), R_);

  bool isfg = t < fgc;
  int keep;
  if (isfg) keep = fgtop[b * R_ + t];
  else      keep = bgtop[b * R_ + (t - fgc) % bga];

  float ex1, ey1, ex2, ey2;
  if (keep < N_) {
    const float* p = rois + ((size_t)b * N_ + keep) * 5;
    ex1 = p[1]; ey1 = p[2]; ex2 = p[3]; ey2 = p[4];
  } else {
    const float* p = gt + ((size_t)b * G_ + (keep - N_)) * 5;
    ex1 = p[0]; ey1 = p[1]; ex2 = p[2]; ey2 = p[3];
  }

  int g = asgn[(size_t)b * M_ + keep];
  const float* gp = gt + ((size_t)b * G_ + g) * 5;
  float lab = isfg ? gp[4] : 0.f;

  float ew = ex2 - ex1 + 1.f, eh = ey2 - ey1 + 1.f;
  float ecx = ex1 + 0.5f * ew, ecy = ey1 + 0.5f * eh;
  float gw = gp[2] - gp[0] + 1.f, gh = gp[3] - gp[1] + 1.f;
  float gcx = gp[0] + 0.5f * gw, gcy = gp[1] + 0.5f * gh;

  float t0 = (gcx - ecx) / ew * 10.f;   // (x - MEAN)/STD, STD=0.1
  float t1 = (gcy - ecy) / eh * 10.f;
  float t2 = logf(gw / ew) * 5.f;       // STD=0.2
  float t3 = logf(gh / eh) * 5.f;
  float fg4 = (lab > 0.f) ? 1.f : 0.f;

  size_t ro = (size_t)b * R_ + t;
  // rois [B,R,5]
  out[ro * 5 + 0] = (float)b;
  out[ro * 5 + 1] = ex1; out[ro * 5 + 2] = ey1;
  out[ro * 5 + 3] = ex2; out[ro * 5 + 4] = ey2;
  // labels [B,R]
  out[20480 + ro] = lab;
  // bbox_targets / inside_w / outside_w [B,R,4]
  float* tg = out + 24576;
  tg[ro * 4 + 0] = t0 * fg4; tg[ro * 4 + 1] = t1 * fg4;
  tg[ro * 4 + 2] = t2 * fg4; tg[ro * 4 + 3] = t3 * fg4;
  float* iw = out + 40960;
  iw[ro * 4 + 0] = fg4; iw[ro * 4 + 1] = fg4;
  iw[ro * 4 + 2] = fg4; iw[ro * 4 + 3] = fg4;
  float* ow = out + 57344;
  ow[ro * 4 + 0] = fg4; ow[ro * 4 + 1] = fg4;
  ow[ro * 4 + 2] = fg4; ow[ro * 4 + 3] = fg4;
}

// ======================================================================
extern "C" void kernel_launch(void* const* d_in, const int* in_sizes, int n_in,
                              void* d_out, int out_size, void* d_ws, size_t ws_size,
                              hipStream_t stream) {
  const float* all_rois = (const float*)d_in[0];   // [16,12000,5] f32
  const float* gt_boxes = (const float*)d_in[1];   // [16,128,5]   f32
  float* out = (float*)d_out;

  // workspace layout (~5.8 MB total)
  char* ws = (char*)d_ws;
  size_t off = 0;
  float* maxov = (float*)(ws + off); off += (size_t)B_ * M_ * sizeof(float);
  int*   asgn  = (int*)  (ws + off); off += (size_t)B_ * M_ * sizeof(int);
  off = (off + 7) & ~(size_t)7;
  u64* keyfg = (u64*)(ws + off); off += (size_t)B_ * NPAD * sizeof(u64);
  u64* keybg = (u64*)(ws + off); off += (size_t)B_ * NPAD * sizeof(u64);
  int* fgnum = (int*)(ws + off); off += (size_t)B_ * sizeof(int);
  int* bgnum = (int*)(ws + off); off += (size_t)B_ * sizeof(int);
  int* fgtop = (int*)(ws + off); off += (size_t)B_ * R_ * sizeof(int);
  int* bgtop = (int*)(ws + off); off += (size_t)B_ * R_ * sizeof(int);
  (void)ws_size; (void)in_sizes; (void)n_in; (void)out_size;

  // allow 128KB dynamic LDS for the sort kernel (host-side, capture-safe)
  (void)hipFuncSetAttribute((const void*)k_sort,
                            hipFuncAttributeMaxDynamicSharedMemorySize,
                            NPAD * (int)sizeof(u64));

  k_zero<<<1, 64, 0, stream>>>(fgnum, 2 * B_);   // fgnum & bgnum contiguous

  dim3 g1((M_ + 127) / 128, B_);
  k_iou<<<g1, 256, 0, stream>>>(all_rois, gt_boxes, maxov, asgn);

  dim3 g2(NPAD / 256, B_);
  k_keys<<<g2, 256, 0, stream>>>(maxov, keyfg, keybg, fgnum, bgnum);

  dim3 g3(1, B_);
  k_sort<<<g3, 1024, NPAD * sizeof(u64), stream>>>(keyfg, fgtop);
  k_sort<<<g3, 1024, NPAD * sizeof(u64), stream>>>(keybg, bgtop);

  k_gather<<<B_, R_, 0, stream>>>(all_rois, gt_boxes, asgn, fgnum, bgnum,
                                  fgtop, bgtop, out);
}